// OrientationDetector_26405458936257
// MI455X (gfx1250) — compile-verified
//
#include <hip/hip_runtime.h>
#include <math.h>

#define PS 32
#define NB 36
#define WAVES_PER_BLOCK 8
#define THREADS (WAVES_PER_BLOCK * 32)

#define PI_F      3.14159265358979323846f
#define HALFPI_F  1.57079632679489661923f

__global__ __launch_bounds__(THREADS) void OrientationDetector_kernel(
    const float* __restrict__ x, const float* __restrict__ gk,
    float* __restrict__ out, int numPatches)
{
    // LDS: shared Gaussian kernel + per-wave patch + per-wave histogram
    __shared__ __attribute__((aligned(16))) float s_gk[PS * PS];
    __shared__ __attribute__((aligned(16))) float s_patch[WAVES_PER_BLOCK][PS * PS];
    __shared__ __attribute__((aligned(16))) float s_hist[WAVES_PER_BLOCK][40]; // 36 used

    const int tid  = threadIdx.x;
    const int wave = tid >> 5;   // wave32 (gfx1250)
    const int lane = tid & 31;

    // Stage gk (4KB) into LDS once per block, 16B per thread (b128 path).
    {
        const float4* g4 = (const float4*)gk;
        float4*       s4 = (float4*)s_gk;
        for (int i = tid; i < (PS * PS) / 4; i += THREADS) s4[i] = g4[i];
    }

    const int p = blockIdx.x * WAVES_PER_BLOCK + wave;  // wave-uniform

    // Zero this wave's histogram.
    for (int i = lane; i < 40; i += 32) s_hist[wave][i] = 0.0f;

    if (p < numPatches) {
        // ---- CDNA5 async copy: patch (4KB) global -> LDS, 8 x b128 per lane ----
        unsigned la = (unsigned)(uintptr_t)(&s_patch[wave][0]) + (unsigned)(lane * 16);
        unsigned long long ga =
            (unsigned long long)(uintptr_t)(x + (size_t)p * (PS * PS)) +
            (unsigned long long)(lane * 16);
#pragma unroll
        for (int k = 0; k < 8; ++k) {
            // vdst = LDS byte address (VGPR), vaddr = 64-bit global address (VGPR pair)
            asm volatile("global_load_async_to_lds_b128 %0, %1, off"
                         :: "v"(la), "v"(ga) : "memory");
            la += 512;
            ga += 512ull;
        }
    }
    // Wait for this wave's async LDS writes (ASYNCcnt).
    asm volatile("s_wait_asynccnt 0" ::: "memory");

    // Block barrier: s_gk visible to all waves (all waves execute this).
    __syncthreads();

    if (p < numPatches) {
        const float* sp   = s_patch[wave];
        float*       hist = s_hist[wave];

        const int lm = (lane == 0)      ? 0      : lane - 1;   // replicate left
        const int lp = (lane == PS - 1) ? PS - 1 : lane + 1;   // replicate right

        // Rolling 3-row window: lane owns column `lane`.
        float curv  = sp[lane];   // row 0
        float prevv = curv;       // replicate top edge

#pragma unroll 8
        for (int r = 0; r < PS; ++r) {
            const float nextv = (r < PS - 1) ? sp[(r + 1) * PS + lane] : curv; // replicate bottom
            const float xl = sp[r * PS + lm];
            const float xr = sp[r * PS + lp];

            const float gxv = 0.5f * (xl - xr);
            const float gyv = 0.5f * (prevv - nextv);

            const float mag =
                __builtin_amdgcn_sqrtf(__builtin_fmaf(gxv, gxv,
                                        __builtin_fmaf(gyv, gyv, 1e-10f)))
                * s_gk[r * PS + lane];

            // ---- fast branchless atan2(gyv, gxv), ~1e-6 rad ----
            const float ax = __builtin_fabsf(gxv);
            const float ay = __builtin_fabsf(gyv);
            const float mx = __builtin_fmaxf(ax, ay);
            const float mn = __builtin_fminf(ax, ay);
            float a = mn * __builtin_amdgcn_rcpf(mx);    // [0,1]; NaN if mx==0
            a = (mx == 0.0f) ? 0.0f : a;                 // atan2(0,0) = 0
            const float s2 = a * a;
            float t = -0.01172120f;
            t = __builtin_fmaf(t, s2,  0.05265332f);
            t = __builtin_fmaf(t, s2, -0.11643287f);
            t = __builtin_fmaf(t, s2,  0.19354346f);
            t = __builtin_fmaf(t, s2, -0.33262347f);
            t = __builtin_fmaf(t, s2,  0.99997726f);
            t = a * t;                                   // atan(a) in [0, pi/4]
            t = (ay > ax)    ? (HALFPI_F - t) : t;       // octant unfold
            t = (gxv < 0.0f) ? (PI_F - t)     : t;
            t = (gyv < 0.0f) ? -t             : t;       // ori in (-pi, pi]

            // o_big = NB * (ori + pi) / (2*pi); reference association keeps o >= 0.
            const float o   = (t + PI_F) * ((float)NB / (2.0f * PI_F));
            const float bo0 = __builtin_floorf(o);
            const float wo1 = o - bo0;
            int b = (int)bo0;
            b = (b >= NB) ? (b - NB) : b;   // ori == pi -> o == 36 -> bin 0 (wo1 == 0)
            b = (b < 0) ? 0 : b;            // safety

            atomicAdd(&hist[b], (1.0f - wo1) * mag);     // ds_add_f32

            prevv = curv;
            curv  = nextv;
        }

        // LDS per-wave in-order; make the RAW on the histogram explicit.
        asm volatile("s_wait_dscnt 0" ::: "memory");

        // ---- zero-padded 3-tap smoothing + argmax over 36 bins ----
        const float inv_hw = 1.0f / (float)(PS * PS);

        const int i1 = lane;                        // bins 0..31
        float v1 = 0.33f * ((i1 > 0) ? hist[i1 - 1] : 0.0f)
                 + 0.34f * hist[i1]
                 + 0.33f * ((i1 < NB - 1) ? hist[i1 + 1] : 0.0f);
        v1 *= inv_hw;

        const int i2 = lane + 32;                   // bins 32..35 on lanes 0..3
        float v2 = -1.0f;                           // hist >= 0, sentinel safe
        if (i2 < NB) {
            v2 = 0.33f * hist[i2 - 1]
               + 0.34f * hist[i2]
               + 0.33f * ((i2 < NB - 1) ? hist[i2 + 1] : 0.0f);
            v2 *= inv_hw;
        }

        // Local best, first-index tie-break (i1 < i2 -> strict >).
        float bv = (v2 > v1) ? v2 : v1;
        int   bi = (v2 > v1) ? i2 : i1;

        // Wave32 butterfly argmax, lowest index wins ties (jnp.argmax).
#pragma unroll
        for (int off = 16; off > 0; off >>= 1) {
            const float ov = __shfl_xor(bv, off, 32);
            const int   oi = __shfl_xor(bi, off, 32);
            if (ov > bv || (ov == bv && oi < bi)) { bv = ov; bi = oi; }
        }

        if (lane == 0) {
            out[p] = -(2.0f * PI_F * (float)bi / (float)NB - PI_F);
        }
    }
}

extern "C" void kernel_launch(void* const* d_in, const int* in_sizes, int n_in,
                              void* d_out, int out_size, void* d_ws, size_t ws_size,
                              hipStream_t stream) {
    const float* x  = (const float*)d_in[0];   // [B,1,32,32] f32
    const float* gk = (const float*)d_in[1];   // [32,32] f32
    float* out = (float*)d_out;                // [B] f32

    const int numPatches = in_sizes[0] / (PS * PS);
    const int blocks = (numPatches + WAVES_PER_BLOCK - 1) / WAVES_PER_BLOCK;

    OrientationDetector_kernel<<<blocks, THREADS, 0, stream>>>(x, gk, out, numPatches);
}